// SplineActivation_71339406787211
// MI455X (gfx1250) — compile-verified
//
#include <hip/hip_runtime.h>

// ============================================================================
// Clamped cubic B-spline activation, KNOTS = linspace(-3,3,16), DEGREE = 3.
//
// Compile-time reduction: for interval j0 = k-3 (k in [3,17], 15 intervals)
// the 4 active basis functions N_{k-3+i}(x) are fixed cubic polynomials in
// u = x - t[k].  We build that 15x4x4 table with a constexpr symbolic
// Cox-de Boor (double precision), then pre-contract it with each channel's
// 18 control coefficients into per-(channel, interval) Horner coefficients:
//     A[w][j0][p] = sum_i coefs[w][j0+i] * P[j0][i][p]
// Hot loop per point: interval index (1 mul + 1 cvt + clamp), one
// ds_load_b128 of float4 A, 3 FMAs, NT store.  Memory bound: 64 MB HBM
// traffic -> ~2.8 us floor at 23.3 TB/s.
// ============================================================================

struct PTab { float v[15][4][4]; };

constexpr double knot_t(int i) {
  int m = i - 3;
  if (m < 0) m = 0;
  if (m > 15) m = 15;
  return -3.0 + 0.4 * (double)m;
}

constexpr PTab make_ptab() {
  PTab P{};
  for (int j0 = 0; j0 < 15; ++j0) {
    const int k = j0 + 3;
    double N[4][4] = {};          // N[i][p]: coeff of u^p of i-th active basis
    N[0][0] = 1.0;
    for (int j = 1; j <= 3; ++j) {
      double saved[4] = {};
      double nN[4][4] = {};
      for (int r = 0; r < j; ++r) {
        const double denom = knot_t(k + 1 + r) - knot_t(k + 1 + r - j);
        double temp[4] = {};
        for (int p = 0; p < 4; ++p) temp[p] = N[r][p] / denom;
        const double a = knot_t(k + 1 + r) - knot_t(k);        // right(u) = a - u
        const double b = knot_t(k) - knot_t(k - (j - 1 - r));  // left(u)  = u + b
        for (int p = 0; p < 4; ++p) {
          const double tm1 = (p > 0) ? temp[p - 1] : 0.0;
          nN[r][p] = saved[p] + a * temp[p] - tm1;
        }
        for (int p = 0; p < 4; ++p) {
          const double tm1 = (p > 0) ? temp[p - 1] : 0.0;
          saved[p] = b * temp[p] + tm1;
        }
      }
      for (int p = 0; p < 4; ++p) nN[j][p] = saved[p];
      for (int q = 0; q <= j; ++q)
        for (int p = 0; p < 4; ++p) N[q][p] = nN[q][p];
    }
    for (int i = 0; i < 4; ++i)
      for (int p = 0; p < 4; ++p) P.v[j0][i][p] = (float)N[i][p];
  }
  return P;
}

constexpr PTab c_P = make_ptab();

#define NCOEF 18          // len(KNOTS) + DEGREE - 1
#define NINT 15           // evaluation intervals
#define CH_TILE 256       // channels per block
#define ROWS_PER_BLOCK 64
#define A_TILE_F4 (NINT * CH_TILE)   // 3840 float4 = 60 KB per tile

// -------- CDNA5 async global->LDS path (guarded; fallback = plain copy) ----
#if defined(__has_builtin)
#if __has_builtin(__builtin_amdgcn_global_load_async_to_lds_b128)
#define HAVE_ASYNC_LDS 1
#endif
#endif
#ifndef HAVE_ASYNC_LDS
#define HAVE_ASYNC_LDS 0
#endif

typedef int v4i __attribute__((vector_size(16)));

// Contract control coefficients of channel w against the constexpr basis
// table; write 15 float4 Horner-coefficient vectors to dst[j0*CH_TILE + ch].
__device__ __forceinline__ void compute_A_channel(const float* __restrict__ coefs,
                                                  int w, int ch,
                                                  float4* __restrict__ dst) {
  float cf[NCOEF];
#pragma unroll
  for (int i = 0; i < NCOEF; ++i) cf[i] = coefs[w * NCOEF + i];
#pragma unroll
  for (int j0 = 0; j0 < NINT; ++j0) {
    float a0 = 0.f, a1 = 0.f, a2 = 0.f, a3 = 0.f;
#pragma unroll
    for (int i = 0; i < 4; ++i) {
      const float c = cf[j0 + i];
      a0 = fmaf(c, c_P.v[j0][i][0], a0);
      a1 = fmaf(c, c_P.v[j0][i][1], a1);
      a2 = fmaf(c, c_P.v[j0][i][2], a2);
      a3 = fmaf(c, c_P.v[j0][i][3], a3);
    }
    dst[j0 * CH_TILE + ch] = make_float4(a0, a1, a2, a3);
  }
}

// Hot loop: one column (channel) per thread, streaming rows.
__device__ __forceinline__ void spline_rows(const float* __restrict__ X,
                                            float* __restrict__ out,
                                            const float4* __restrict__ Alds4,
                                            int W, int col, int ch,
                                            int b0, int b1, int B) {
#pragma unroll 4
  for (int b = b0; b < b1; ++b) {
    const int pb = (b + 16 < B) ? (b + 16) : (B - 1);
    __builtin_prefetch(X + (size_t)pb * W + col, 0, 0);      // global_prefetch_b8
    const float x = X[(size_t)b * W + col];
    const float xp3 = x + 3.0f;
    int j = (int)(xp3 * 2.5f);                   // trunc == floor after clamp
    j = (j < 0) ? 0 : ((j > NINT - 1) ? (NINT - 1) : j);
    const float u = fmaf(-0.4f, (float)j, xp3);  // u = x - t[k]
    const float4 a = Alds4[j * CH_TILE + ch];    // ds_load_b128, conflict-free
    const float y = fmaf(fmaf(fmaf(a.w, u, a.z), u, a.y), u, a.x);
    __builtin_nontemporal_store(y, out + (size_t)b * W + col);
  }
}

// Kernel 1: contract coefs -> workspace A table (tiny: 61K FMAs per block).
__global__ void __launch_bounds__(CH_TILE)
spline_precomp_kernel(const float* __restrict__ coefs, float4* __restrict__ wsA) {
  const int ch = threadIdx.x;
  const int tile = blockIdx.x;
  compute_A_channel(coefs, tile * CH_TILE + ch, ch, wsA + (size_t)tile * A_TILE_F4);
}

// Kernel 2a: stage this tile's A into LDS via async b128 copy, then stream.
__global__ void __launch_bounds__(CH_TILE)
spline_main_staged(const float* __restrict__ X, const float* __restrict__ wsA,
                   float* __restrict__ out, int B, int W, int rowsPerBlock) {
  __shared__ float4 Alds4[A_TILE_F4];            // 60 KB of 320 KB/WGP
  const int ch = threadIdx.x;
  const int tile = blockIdx.x;
  const float4* src4 = (const float4*)(wsA) + (size_t)tile * A_TILE_F4;

#if HAVE_ASYNC_LDS
#pragma unroll
  for (int it = 0; it < NINT; ++it) {
    const int idx = it * CH_TILE + ch;
    __builtin_amdgcn_global_load_async_to_lds_b128(
        (v4i*)(src4 + idx), (v4i*)(Alds4 + idx), 0, 0);
  }
#if defined(__has_builtin) && __has_builtin(__builtin_amdgcn_s_wait_asynccnt)
  __builtin_amdgcn_s_wait_asynccnt(0);
#else
  asm volatile("s_wait_asynccnt 0" ::: "memory");
#endif
#else
#pragma unroll
  for (int it = 0; it < NINT; ++it) {
    const int idx = it * CH_TILE + ch;
    Alds4[idx] = src4[idx];
  }
#endif
  __syncthreads();

  const int b0 = blockIdx.y * rowsPerBlock;
  int b1 = b0 + rowsPerBlock;
  if (b1 > B) b1 = B;
  spline_rows(X, out, Alds4, W, tile * CH_TILE + ch, ch, b0, b1, B);
}

// Kernel 2b: fallback if workspace is too small — compute A in-block.
__global__ void __launch_bounds__(CH_TILE)
spline_main_fused(const float* __restrict__ X, const float* __restrict__ coefs,
                  float* __restrict__ out, int B, int W, int rowsPerBlock) {
  __shared__ float4 Alds4[A_TILE_F4];
  const int ch = threadIdx.x;
  const int tile = blockIdx.x;
  compute_A_channel(coefs, tile * CH_TILE + ch, ch, Alds4);
  __syncthreads();

  const int b0 = blockIdx.y * rowsPerBlock;
  int b1 = b0 + rowsPerBlock;
  if (b1 > B) b1 = B;
  spline_rows(X, out, Alds4, W, tile * CH_TILE + ch, ch, b0, b1, B);
}

extern "C" void kernel_launch(void* const* d_in, const int* in_sizes, int n_in,
                              void* d_out, int out_size, void* d_ws, size_t ws_size,
                              hipStream_t stream) {
  const float* X = (const float*)d_in[0];      // [B, W] fp32
  const float* coefs = (const float*)d_in[1];  // [W, 18] fp32
  float* out = (float*)d_out;                  // [B, W] fp32

  const int W = in_sizes[1] / NCOEF;           // 1024
  const int B = in_sizes[0] / W;               // 8192
  const int tiles = W / CH_TILE;               // 4
  const int rowsPerBlock = ROWS_PER_BLOCK;
  const int ychunks = (B + rowsPerBlock - 1) / rowsPerBlock;

  const size_t needWS = (size_t)tiles * A_TILE_F4 * sizeof(float4);
  if (ws_size >= needWS) {
    spline_precomp_kernel<<<dim3(tiles), dim3(CH_TILE), 0, stream>>>(
        coefs, (float4*)d_ws);
    spline_main_staged<<<dim3(tiles, ychunks), dim3(CH_TILE), 0, stream>>>(
        X, (const float*)d_ws, out, B, W, rowsPerBlock);
  } else {
    spline_main_fused<<<dim3(tiles, ychunks), dim3(CH_TILE), 0, stream>>>(
        X, coefs, out, B, W, rowsPerBlock);
  }
}